// BinaryOneToManyMatcher_24670292148869
// MI455X (gfx1250) — compile-verified
//
#include <hip/hip_runtime.h>
#include <cstdint>
#include <cstddef>

// Problem constants (match setup_inputs in the reference).
#define B_  128
#define Q_  1000
#define G_  300
#define K_  4
#define THRESH 0.4f

typedef unsigned int u32x4 __attribute__((ext_vector_type(4)));
typedef int          i32x4 __attribute__((ext_vector_type(4)));
typedef int          i32x8 __attribute__((ext_vector_type(8)));

#if __has_builtin(__builtin_amdgcn_tensor_load_to_lds) && __has_builtin(__builtin_amdgcn_s_wait_tensorcnt)
#define HAVE_TDM 1
#else
#define HAVE_TDM 0
#endif

#if HAVE_TDM
// Issue a TDM load of `n` contiguous 32-bit elements from global `src` into
// LDS byte offset `lds_byte`. D# layout per CDNA5 ISA sec 8.3/8.4:
//   group0: [1:0]=count=1, [63:32]=lds_addr, [120:64]=global_addr, [127:126]=type=2
//   group1: data_size=4B, tensor_dim0=n, tensor_dim1=1, tile_dim0=n, tile_dim1=1,
//           tensor_dim0_stride=n, everything else 0.
__device__ __forceinline__ void tdm_load_f32_1d(const float* src, uint32_t lds_byte, uint32_t n) {
  uint64_t ga = (uint64_t)(uintptr_t)src;
  u32x4 g0;
  g0[0] = 1u;                                              // count=1 (valid), no gather
  g0[1] = lds_byte;                                        // lds_addr
  g0[2] = (uint32_t)ga;                                    // global_addr[31:0]
  g0[3] = (uint32_t)((ga >> 32) & 0x01FFFFFFu) | (2u << 30); // global_addr[56:32] | type=2
  i32x8 g1;
  g1[0] = (int)(2u << 16);                                 // data_size = 2 (4 bytes)
  g1[1] = (int)((n & 0xFFFFu) << 16);                      // tensor_dim0[15:0] @ bits 63:48
  g1[2] = (int)(((n >> 16) & 0xFFFFu) | (1u << 16));       // tensor_dim0[31:16] | tensor_dim1=1
  g1[3] = (int)((n & 0xFFFFu) << 16);                      // tensor_dim1[31:16]=0 | tile_dim0=n
  g1[4] = 1;                                               // tile_dim1=1, tile_dim2=0
  g1[5] = (int)n;                                          // tensor_dim0_stride[31:0]
  g1[6] = 0;                                               // stride0[47:32]=0, stride1[15:0]=0
  g1[7] = 0;                                               // stride1[47:16]=0
  i32x4 z4 = {0, 0, 0, 0};
#if __clang_major__ >= 23
  i32x8 z8 = {0, 0, 0, 0, 0, 0, 0, 0};
  __builtin_amdgcn_tensor_load_to_lds(g0, g1, z4, z4, z8, 0);
#else
  __builtin_amdgcn_tensor_load_to_lds(g0, g1, z4, z4, 0);
#endif
}
#endif

__global__ __launch_bounds__(128)
void binary_matcher_kernel(const float* __restrict__ logits,      // [B,Q,1]
                           const float* __restrict__ pboxes,      // [B,Q,4]
                           const float* __restrict__ tboxes,      // [B,G,4]
                           float* __restrict__ out_vals,          // [B,G,K] f32
                           int*   __restrict__ out_idxs,          // [B,G,K] i32
                           float* __restrict__ out_mask) {        // [B,G,K] 0/1
  __shared__ __align__(16) float s_boxes[Q_ * 4];   // 16000 B
  __shared__ float s_score[Q_];                     //  4000 B (logits, then sigmoid in place)
  __shared__ float s_area[Q_];                      //  4000 B

  const int b   = blockIdx.y;
  const int tid = threadIdx.x;
  const float* pb_b = pboxes + (size_t)b * (Q_ * 4);
  const float* lg_b = logits + (size_t)b * Q_;

#if HAVE_TDM
  // Two TDM issues per block (EXEC is ignored by TDM; lane-0 branch keeps it to
  // wave 0). Other waves see TENSORcnt==0 and fall through the wait.
  if (tid == 0) {
    tdm_load_f32_1d(pb_b, (uint32_t)(uintptr_t)&s_boxes[0], Q_ * 4);
    tdm_load_f32_1d(lg_b, (uint32_t)(uintptr_t)&s_score[0], Q_);
  }
  __builtin_amdgcn_s_wait_tensorcnt(0);
#else
  for (int i = tid; i < Q_ * 4; i += blockDim.x) s_boxes[i] = pb_b[i];
  for (int q = tid; q < Q_;     q += blockDim.x) s_score[q] = lg_b[q];
#endif
  __syncthreads();

  // Block-cooperative: sigmoid(logit) in place + per-query box area.
  for (int q = tid; q < Q_; q += blockDim.x) {
    float lgv  = s_score[q];
    s_score[q] = 1.0f / (1.0f + __expf(-lgv));
    float x1 = s_boxes[q * 4 + 0], y1 = s_boxes[q * 4 + 1];
    float x2 = s_boxes[q * 4 + 2], y2 = s_boxes[q * 4 + 3];
    s_area[q] = (x2 - x1) * (y2 - y1);
  }
  __syncthreads();

  const int g = blockIdx.x * blockDim.x + tid;
  if (g >= G_) return;

  const float4 tb = *reinterpret_cast<const float4*>(tboxes + ((size_t)b * G_ + g) * 4);
  // Fold the reference's +1e-7 union epsilon into the per-lane GT area.
  const float a2e = (tb.z - tb.x) * (tb.w - tb.y) + 1e-7f;

  // Sorted top-4 (descending). Init -1 so the first 4 zero-candidates fill
  // indices 0..3, matching jax.lax.top_k's stable tie behavior.
  float v0 = -1.0f, v1 = -1.0f, v2 = -1.0f, v3 = -1.0f;
  int   i0 = 0, i1 = 0, i2 = 0, i3 = 0;

  for (int q = 0; q < Q_; ++q) {
    // All lanes of a wave read the same LDS address -> broadcast, no conflicts.
    const float4 pbq = *reinterpret_cast<const float4*>(&s_boxes[q * 4]);
    float ltx = fmaxf(pbq.x, tb.x);
    float lty = fmaxf(pbq.y, tb.y);
    float rbx = fminf(pbq.z, tb.z);
    float rby = fminf(pbq.w, tb.w);
    float w = fmaxf(rbx - ltx, 0.0f);
    float h = fmaxf(rby - lty, 0.0f);
    float inter = w * h;
    float denom = (s_area[q] + a2e) - inter;               // union + 1e-7
    float iou   = inter * __builtin_amdgcn_rcpf(denom);
    float s = (iou > THRESH) ? s_score[q] * iou : 0.0f;
    if (s > v3) {                                          // rare in practice
      if (s > v0)      { v3=v2; i3=i2; v2=v1; i2=i1; v1=v0; i1=i0; v0=s; i0=q; }
      else if (s > v1) { v3=v2; i3=i2; v2=v1; i2=i1; v1=s; i1=q; }
      else if (s > v2) { v3=v2; i3=i2; v2=s; i2=q; }
      else             { v3=s; i3=q; }
    }
  }

  const int base = ((b * G_) + g) * K_;
  out_vals[base + 0] = v0; out_vals[base + 1] = v1;
  out_vals[base + 2] = v2; out_vals[base + 3] = v3;
  out_idxs[base + 0] = i0; out_idxs[base + 1] = i1;
  out_idxs[base + 2] = i2; out_idxs[base + 3] = i3;
  out_mask[base + 0] = (v0 > 0.0f) ? 1.0f : 0.0f;
  out_mask[base + 1] = (v1 > 0.0f) ? 1.0f : 0.0f;
  out_mask[base + 2] = (v2 > 0.0f) ? 1.0f : 0.0f;
  out_mask[base + 3] = (v3 > 0.0f) ? 1.0f : 0.0f;
}

extern "C" void kernel_launch(void* const* d_in, const int* in_sizes, int n_in,
                              void* d_out, int out_size, void* d_ws, size_t ws_size,
                              hipStream_t stream) {
  (void)in_sizes; (void)n_in; (void)out_size; (void)d_ws; (void)ws_size;
  const float* logits = (const float*)d_in[0];   // [B,Q,1] f32
  const float* pboxes = (const float*)d_in[1];   // [B,Q,4] f32
  const float* tboxes = (const float*)d_in[2];   // [B,G,4] f32

  const size_t bgk = (size_t)B_ * G_ * K_;
  float* vals = (float*)d_out;                   // output 0: vals  [B,G,K] f32
  int*   idxs = (int*)((float*)d_out + bgk);     // output 1: idxs  [B,G,K] i32
  float* mask = (float*)d_out + 2 * bgk;         // output 2: mask  [B,G,K] 0/1

  dim3 grid((G_ + 127) / 128, B_);
  dim3 block(128);
  binary_matcher_kernel<<<grid, block, 0, stream>>>(logits, pboxes, tboxes, vals, idxs, mask);
}